// MLPTopK_BN_1400159339075
// MI455X (gfx1250) — compile-verified
//
#include <hip/hip_runtime.h>
#include <hip/hip_bf16.h>
#include <cstdint>

typedef __attribute__((ext_vector_type(2))) float v2f;
typedef __attribute__((ext_vector_type(8))) float v8f;

#define BATCH 8
#define NPTS  8192
#define CIN   128
#define COUT  256
#define MSEL  2048
#define KNN   16
#define ROWS  (BATCH * NPTS)   // 65536
#define BN_EPS 1e-5f

// ---------------------------------------------------------------------------
// gfx1250 async global->LDS copy (16B per lane-op) via inline CDNA5 asm,
// with portable fallback. Tracked by ASYNCcnt (s_wait_asynccnt).
// ---------------------------------------------------------------------------
#if defined(__gfx1250__)
#define ASYNC_LDS 1
#else
#define ASYNC_LDS 0
#endif

__device__ __forceinline__ void copy16_g2l(const float* g, float* l) {
#if ASYNC_LDS
    // VDST = LDS byte address (low 32 bits of generic pointer), VADDR = 64-bit
    // global address, SADDR = off.  ISA: GLOBAL_LOAD_ASYNC_TO_LDS_B128.
    const uint32_t lds_addr = (uint32_t)(uintptr_t)l;
    asm volatile("global_load_async_to_lds_b128 %0, %1, off"
                 :
                 : "v"(lds_addr), "v"(g)
                 : "memory");
#else
    *(float4*)l = *(const float4*)g;
#endif
}

__device__ __forceinline__ void async_copy_wait() {
#if ASYNC_LDS
    asm volatile("s_wait_asynccnt 0x0" ::: "memory");
#endif
}

// ---------------------------------------------------------------------------
// GEMM + BN + ReLU:  Out[r][c] = relu(bn( sum_k A[r][k] * W[c][k] ))
// A: (ROWS x K) row-major, W: (COUT x K) row-major, Out: (ROWS x COUT)
// Per wave: 16 rows x 64 cols via v_wmma_f32_16x16x4_f32, K-step = 4.
// Block = 128 threads = 4 waves; the 64-col W panel is staged once per block
// into LDS (async), since all 4 waves consume identical B fragments.
// LDS row stride K+4 floats: keeps 16B alignment and spreads the 16 lanes of
// a B-fragment read across banks 4m..4m+1 (conflict-free).
// ---------------------------------------------------------------------------
template <int K>
__global__ __launch_bounds__(128)
void gemm_bn_relu_kernel(const float* __restrict__ A,
                         const float* __restrict__ W,
                         const float* __restrict__ gamma,
                         const float* __restrict__ beta,
                         const float* __restrict__ mean,
                         const float* __restrict__ var,
                         float* __restrict__ Out)
{
    constexpr int KP = K + 4;
    __shared__ float lds_w[64 * KP];

    const int lane    = threadIdx.x & 31;
    const int wave    = threadIdx.x >> 5;
    const int rowBase = (blockIdx.x * 4 + wave) * 16;
    const int colBase = blockIdx.y * 64;

    // Stage W[colBase .. colBase+63][0..K) into LDS.
    constexpr int CHUNKS = 64 * (K / 4);
    for (int c4 = threadIdx.x; c4 < CHUNKS; c4 += 128) {
        const int row = c4 / (K / 4);
        const int pos = (c4 % (K / 4)) * 4;
        copy16_g2l(W + (size_t)(colBase + row) * K + pos, &lds_w[row * KP + pos]);
    }
    async_copy_wait();
    __syncthreads();

    const int m  = lane & 15;          // row within A-frag / col within B-frag
    const int kh = (lane >> 4) * 2;    // K sub-offset per lane half

    const float* arow = A + (size_t)(rowBase + m) * K;

    v8f acc[4];
    acc[0] = v8f{}; acc[1] = v8f{}; acc[2] = v8f{}; acc[3] = v8f{};

    for (int k = 0; k < K; k += 4) {
        // A fragment: A[rowBase+m][k+kh .. k+kh+1]  (global stream)
        v2f a = *(const v2f*)(arow + k + kh);
        // B fragments from LDS: B[kk][n] = W[colBase+16t+n][k+kk]
        v2f b0 = *(const v2f*)&lds_w[(m +  0) * KP + k + kh];
        v2f b1 = *(const v2f*)&lds_w[(m + 16) * KP + k + kh];
        v2f b2 = *(const v2f*)&lds_w[(m + 32) * KP + k + kh];
        v2f b3 = *(const v2f*)&lds_w[(m + 48) * KP + k + kh];
        acc[0] = __builtin_amdgcn_wmma_f32_16x16x4_f32(false, a, false, b0, 0, acc[0], false, false);
        acc[1] = __builtin_amdgcn_wmma_f32_16x16x4_f32(false, a, false, b1, 0, acc[1], false, false);
        acc[2] = __builtin_amdgcn_wmma_f32_16x16x4_f32(false, a, false, b2, 0, acc[2], false, false);
        acc[3] = __builtin_amdgcn_wmma_f32_16x16x4_f32(false, a, false, b3, 0, acc[3], false, false);
    }

    // Epilogue: C/D layout -> row = rowBase + v + 8*(lane>>4), col = colBase+16t+(lane&15)
    const int rbase = rowBase + 8 * (lane >> 4);
    #pragma unroll
    for (int t = 0; t < 4; ++t) {
        const int col = colBase + 16 * t + (lane & 15);
        const float sc = gamma[col] * rsqrtf(var[col] + BN_EPS);
        const float sh = beta[col] - mean[col] * sc;
        #pragma unroll
        for (int v = 0; v < 8; ++v) {
            float val = acc[t][v] * sc + sh;
            val = val > 0.0f ? val : 0.0f;
            Out[(size_t)(rbase + v) * COUT + col] = val;
        }
    }
}

// ---------------------------------------------------------------------------
// Scores: one wave per row; score[r] = dot(F[r][:], Ws) + bs
// ---------------------------------------------------------------------------
__global__ __launch_bounds__(256)
void score_kernel(const float* __restrict__ F,
                  const float* __restrict__ Ws,
                  const float* __restrict__ bs,
                  float* __restrict__ scores)
{
    const int gwave = (blockIdx.x * 256 + threadIdx.x) >> 5;
    const int lane  = threadIdx.x & 31;
    if (gwave >= ROWS) return;
    const float* row = F + (size_t)gwave * COUT + lane * 8;
    const float* w   = Ws + lane * 8;
    float s = 0.0f;
    #pragma unroll
    for (int i = 0; i < 8; ++i) s += row[i] * w[i];
    #pragma unroll
    for (int off = 16; off > 0; off >>= 1) s += __shfl_xor(s, off, 32);
    if (lane == 0) scores[gwave] = s + bs[0];
}

// ---------------------------------------------------------------------------
// Per-batch descending bitonic sort of (score, idx); emit p_out for top-M.
// One 1024-thread workgroup per batch; 64 KB LDS; scores staged async.
// ---------------------------------------------------------------------------
__global__ __launch_bounds__(1024)
void topm_kernel(const float* __restrict__ scores,
                 const float* __restrict__ p,
                 float* __restrict__ p_out)
{
    __shared__ float key[NPTS];
    __shared__ int   idx[NPTS];
    const int b = blockIdx.x;
    const float* s = scores + (size_t)b * NPTS;

    for (int c4 = threadIdx.x; c4 < NPTS / 4; c4 += 1024)
        copy16_g2l(s + c4 * 4, &key[c4 * 4]);
    for (int i = threadIdx.x; i < NPTS; i += 1024) idx[i] = i;
    async_copy_wait();
    __syncthreads();

    for (int k = 2; k <= NPTS; k <<= 1) {
        for (int j = k >> 1; j > 0; j >>= 1) {
            for (int t = threadIdx.x; t < NPTS / 2; t += 1024) {
                const int i   = ((t & ~(j - 1)) << 1) | (t & (j - 1));
                const int ixj = i | j;
                const bool up = ((i & k) == 0);
                const float a = key[i], c = key[ixj];
                // overall descending sort
                if (up ? (a < c) : (a > c)) {
                    key[i] = c; key[ixj] = a;
                    const int ti = idx[i]; idx[i] = idx[ixj]; idx[ixj] = ti;
                }
            }
            __syncthreads();
        }
    }

    for (int t = threadIdx.x; t < MSEL; t += 1024) {
        const int id = idx[t];
        const float* pp = p + ((size_t)b * NPTS + id) * 3;
        float* po = p_out + ((size_t)b * MSEL + t) * 3;
        po[0] = pp[0]; po[1] = pp[1]; po[2] = pp[2];
    }
}

// ---------------------------------------------------------------------------
// KNN-16: one thread per query; the batch's 8192 points staged in LDS (96 KB,
// AoS — the inner loop reads one point uniformly -> pure LDS broadcast).
// Register-resident ascending 16-best insertion list (set only; order-free
// since downstream is a max-pool).
// ---------------------------------------------------------------------------
__global__ __launch_bounds__(256)
void knn_kernel(const float* __restrict__ p,
                const float* __restrict__ p_out,
                int* __restrict__ nbr)
{
    __shared__ float pl[NPTS * 3];
    const int b = blockIdx.y;
    const float* pb = p + (size_t)b * NPTS * 3;
    for (int c4 = threadIdx.x; c4 < (NPTS * 3) / 4; c4 += 256)
        copy16_g2l(pb + c4 * 4, &pl[c4 * 4]);
    async_copy_wait();
    __syncthreads();

    const int m = blockIdx.x * 256 + threadIdx.x;
    const float* q = p_out + ((size_t)b * MSEL + m) * 3;
    const float qx = q[0], qy = q[1], qz = q[2];

    float bd[KNN];
    int   bi[KNN];
    #pragma unroll
    for (int t = 0; t < KNN; ++t) { bd[t] = 3.4e38f; bi[t] = 0; }

    for (int n = 0; n < NPTS; ++n) {
        const float dx = qx - pl[3 * n + 0];
        const float dy = qy - pl[3 * n + 1];
        const float dz = qz - pl[3 * n + 2];
        const float d = dx * dx + dy * dy + dz * dz;
        if (d < bd[KNN - 1]) {
            float cd = d; int ci = n;
            #pragma unroll
            for (int t = 0; t < KNN; ++t) {
                if (cd < bd[t]) {
                    const float tf = bd[t]; bd[t] = cd; cd = tf;
                    const int   ti = bi[t]; bi[t] = ci; ci = ti;
                }
            }
        }
    }

    int* o = nbr + ((size_t)b * MSEL + m) * KNN;
    #pragma unroll
    for (int t = 0; t < KNN; ++t) o[t] = bi[t];
}

// ---------------------------------------------------------------------------
// Neighbor feature max-pool: one block per (b,m); thread = channel.
// ---------------------------------------------------------------------------
__global__ __launch_bounds__(256)
void featmax_kernel(const float* __restrict__ F,
                    const int* __restrict__ nbr,
                    float* __restrict__ y)
{
    __shared__ int nbs[KNN];
    const int bm = blockIdx.x;           // 0 .. BATCH*MSEL-1
    const int b  = bm / MSEL;
    if (threadIdx.x < KNN) nbs[threadIdx.x] = nbr[(size_t)bm * KNN + threadIdx.x];
    __syncthreads();

    const int c = threadIdx.x;
    const float* Fb = F + (size_t)b * NPTS * COUT;
    float best = -3.4e38f;
    #pragma unroll
    for (int t = 0; t < KNN; ++t) {
        const float v = Fb[(size_t)nbs[t] * COUT + c];
        best = v > best ? v : best;
    }
    y[(size_t)bm * COUT + c] = best;
}

// ---------------------------------------------------------------------------
extern "C" void kernel_launch(void* const* d_in, const int* in_sizes, int n_in,
                              void* d_out, int out_size, void* d_ws, size_t ws_size,
                              hipStream_t stream) {
    (void)in_sizes; (void)n_in; (void)out_size; (void)ws_size;

    const float* x  = (const float*)d_in[0];
    const float* p  = (const float*)d_in[1];
    const float* W1 = (const float*)d_in[2];
    const float* g1 = (const float*)d_in[3];
    const float* b1 = (const float*)d_in[4];
    const float* m1 = (const float*)d_in[5];
    const float* v1 = (const float*)d_in[6];
    const float* W2 = (const float*)d_in[7];
    const float* g2 = (const float*)d_in[8];
    const float* b2 = (const float*)d_in[9];
    const float* m2 = (const float*)d_in[10];
    const float* v2 = (const float*)d_in[11];
    const float* Ws = (const float*)d_in[12];
    const float* bs = (const float*)d_in[13];

    // Workspace layout
    char* ws = (char*)d_ws;
    float* h      = (float*)(ws);                                  // 64 MB
    float* mlp    = (float*)(ws + (size_t)ROWS * COUT * 4);        // 64 MB
    float* scores = (float*)(ws + (size_t)2 * ROWS * COUT * 4);    // 256 KB
    int*   nbr    = (int*)  (ws + (size_t)2 * ROWS * COUT * 4 + (size_t)ROWS * 4);

    // Output layout: y (B*M*COUT) then p_out (B*M*3)
    float* y     = (float*)d_out;
    float* p_out = (float*)d_out + (size_t)BATCH * MSEL * COUT;

    // 1) h = relu(bn(x @ W1^T))
    gemm_bn_relu_kernel<CIN><<<dim3(ROWS / 64, COUT / 64), 128, 0, stream>>>(
        x, W1, g1, b1, m1, v1, h);
    // 2) mlp = relu(bn(h @ W2^T))
    gemm_bn_relu_kernel<COUT><<<dim3(ROWS / 64, COUT / 64), 128, 0, stream>>>(
        h, W2, g2, b2, m2, v2, mlp);
    // 3) scores
    score_kernel<<<ROWS / 8, 256, 0, stream>>>(mlp, Ws, bs, scores);
    // 4) top-M select (descending) + gather p_out
    topm_kernel<<<BATCH, 1024, 0, stream>>>(scores, p, p_out);
    // 5) KNN-16
    knn_kernel<<<dim3(MSEL / 256, BATCH), 256, 0, stream>>>(p, p_out, nbr);
    // 6) neighbor feature max-pool
    featmax_kernel<<<BATCH * MSEL, 256, 0, stream>>>(mlp, nbr, y);
}